// TopoPoolSAPI_70214125355059
// MI455X (gfx1250) — compile-verified
//
#include <hip/hip_runtime.h>
#include <math.h>

// Problem constants (from reference)
#define BGR   64                      // graphs
#define NPG   4096                    // nodes per graph
#define TOTAL (BGR*NPG)               // 262144 nodes
#define EPG   65536
#define ETOT  ((long)BGR*EPG)         // 4194304 edges
#define CDIM  128
#define DTOK  64
#define DATT  64
#define NTOK  8
#define KSEL  2048                    // ceil(0.5*NPG)

typedef __attribute__((ext_vector_type(2))) float v2f;
typedef __attribute__((ext_vector_type(8))) float v8f;

// ---- d_out layout (floats, tuple concatenated in return order) ----
#define X_OUT_OFF 0L
#define X_OUT_LEN ((long)BGR*KSEL*CDIM)          // 16777216
#define EI_OFF    (X_OUT_OFF + X_OUT_LEN)
#define EI_LEN    (2L*ETOT)                      // 8388608
#define BATCH_OFF (EI_OFF + EI_LEN)
#define PERM_OFF  (BATCH_OFF + (long)BGR*KSEL)
#define SCORE_OFF (PERM_OFF + (long)BGR*KSEL)
#define MASK_OFF  (SCORE_OFF + (long)BGR*KSEL)

// ---- workspace layout (element offsets into d_ws) ----
#define WS_C      0         // 1 float : c = sum(w)/||w||
#define WS_V      64        // 8192 floats : V[B][C] = Wq^T Wk t_bar
#define WS_SCORE  8448      // 262144 floats : per-node tanh score
#define WS_PERM   270848    // 131072 ints  : kept global node ids
#define WS_SSEL   401920    // 131072 floats: kept scores
#define WS_MAP    532992    // 262144 ints  : old node -> new id or -1

// ============================================================
// Kernel 0: t_bar mean, scalar c, and V = (t_bar Wk^T) Wq via
// V_WMMA_F32_16X16X4_F32. Single workgroup, 16 waves.
// ============================================================
__global__ __launch_bounds__(512)
void prep_kernel(const float* __restrict__ topo, const float* __restrict__ Wq,
                 const float* __restrict__ Wk, const float* __restrict__ pw,
                 float* __restrict__ ws_f) {
  __shared__ float tbar[BGR * DTOK];   // 64x64 = 16 KB
  __shared__ float kbuf[BGR * DATT];   // 64x64 = 16 KB
  const int tid = threadIdx.x;

  // t_bar[b][d] = mean_t topo[b][t][d]
  for (int i = tid; i < BGR * DTOK; i += 512) {
    int b = i >> 6, d = i & 63;
    const float* p = topo + (long)b * NTOK * DTOK + d;
    float s = 0.f;
#pragma unroll
    for (int t = 0; t < NTOK; ++t) s += p[t * DTOK];
    tbar[i] = s * (1.0f / NTOK);
  }
  // c = sum(pool_w) / ||pool_w||  (wave 0; 128 elems = 32 lanes x float4)
  if (tid < 32) {
    float4 wv = ((const float4*)pw)[tid];
    float s = wv.x + wv.y + wv.z + wv.w;
    float q = wv.x*wv.x + wv.y*wv.y + wv.z*wv.z + wv.w*wv.w;
#pragma unroll
    for (int o = 16; o; o >>= 1) { s += __shfl_xor(s, o, 32); q += __shfl_xor(q, o, 32); }
    if (tid == 0) ws_f[WS_C] = s * rsqrtf(q);
  }
  __syncthreads();

  const int wave = tid >> 5, lane = tid & 31;
  const int hi = lane >> 4, lo = lane & 15;

  // GEMM1: kb[b][a] = sum_d tbar[b][d] * Wk[a][d]   (M=N=K=64, 16 tiles)
  {
    const int tm = wave >> 2, tn = wave & 3;
    v8f acc = {};
#pragma unroll
    for (int kk = 0; kk < DTOK; kk += 4) {
      const int k0 = kk + 2 * hi;
      v2f a, b;
      a[0] = tbar[(tm*16 + lo) * DTOK + k0];
      a[1] = tbar[(tm*16 + lo) * DTOK + k0 + 1];
      b[0] = Wk[(tn*16 + lo) * DTOK + k0];       // B(k,n) = Wk[n][k]
      b[1] = Wk[(tn*16 + lo) * DTOK + k0 + 1];
      acc = __builtin_amdgcn_wmma_f32_16x16x4_f32(false, a, false, b,
                                                  (short)0, acc, false, false);
    }
#pragma unroll
    for (int j = 0; j < 8; ++j)
      kbuf[(tm*16 + j + 8*hi) * DATT + tn*16 + lo] = acc[j];
  }
  __syncthreads();

  // GEMM2: V[b][c] = sum_a kb[b][a] * Wq[a][c]  (M=64,N=128,K=64, 32 tiles)
#pragma unroll
  for (int rep = 0; rep < 2; ++rep) {
    const int t = wave + rep * 16;
    const int tm = t >> 3, tn = t & 7;
    v8f acc = {};
#pragma unroll
    for (int kk = 0; kk < DATT; kk += 4) {
      const int k0 = kk + 2 * hi;
      v2f a, b;
      a[0] = kbuf[(tm*16 + lo) * DATT + k0];
      a[1] = kbuf[(tm*16 + lo) * DATT + k0 + 1];
      b[0] = Wq[(long)k0 * CDIM + tn*16 + lo];   // B(k,n) = Wq[k][n]
      b[1] = Wq[(long)(k0+1) * CDIM + tn*16 + lo];
      acc = __builtin_amdgcn_wmma_f32_16x16x4_f32(false, a, false, b,
                                                  (short)0, acc, false, false);
    }
#pragma unroll
    for (int j = 0; j < 8; ++j)
      ws_f[WS_V + (long)(tm*16 + j + 8*hi) * CDIM + tn*16 + lo] = acc[j];
  }
}

// ============================================================
// Kernel 1: score[i] = tanh( (x[i] . V[b]) * c / sqrt(DATT) )
// wave-per-node: lane l holds float4 chunk -> contiguous 512B row
// ============================================================
__global__ __launch_bounds__(256)
void score_kernel(const float* __restrict__ x, const float* __restrict__ ws_f,
                  float* __restrict__ score) {
  const int wave = threadIdx.x >> 5, lane = threadIdx.x & 31;
  const long node = (long)blockIdx.x * 8 + wave;
  const int b = (int)(node >> 12);  // node / NPG
  float4 xv = ((const float4*)(x + node * CDIM))[lane];
  float4 vv = ((const float4*)(ws_f + WS_V + (long)b * CDIM))[lane];
  float s = xv.x*vv.x + xv.y*vv.y + xv.z*vv.z + xv.w*vv.w;
#pragma unroll
  for (int o = 16; o; o >>= 1) s += __shfl_xor(s, o, 32);
  if (lane == 0) {
    const float c = ws_f[WS_C];
    score[node] = tanhf(s * 0.125f * c);   // 1/sqrt(64) = 0.125
  }
}

// ============================================================
// Kernel 2: per-graph bitonic sort (score desc, index asc) in LDS,
// emit perm/score_sel/batch_out and full node->newid mapping.
// ============================================================
__global__ __launch_bounds__(1024)
void topk_kernel(const float* __restrict__ score, int* __restrict__ mapping,
                 int* __restrict__ perm_ws, float* __restrict__ ssel_ws,
                 float* __restrict__ out) {
  __shared__ float s[NPG];
  __shared__ int   id[NPG];
  const int b = blockIdx.x, tid = threadIdx.x;
  for (int i = tid; i < NPG; i += 1024) { s[i] = score[(long)b * NPG + i]; id[i] = i; }
  __syncthreads();
  for (int kk = 2; kk <= NPG; kk <<= 1) {
    for (int j = kk >> 1; j > 0; j >>= 1) {
      for (int i = tid; i < NPG; i += 1024) {
        const int ixj = i ^ j;
        if (ixj > i) {
          const bool desc = ((i & kk) == 0);
          float si = s[i], sx = s[ixj];
          int   ii = id[i], ix = id[ixj];
          // strict "i comes after ixj" under (score desc, index asc)
          const bool oo = (si < sx) || ((si == sx) && (ii > ix));
          if (oo == desc) { s[i] = sx; s[ixj] = si; id[i] = ix; id[ixj] = ii; }
        }
      }
      __syncthreads();
    }
  }
  for (int i = tid; i < NPG; i += 1024) {
    const int gnode = b * NPG + id[i];
    if (i < KSEL) {
      const int r = b * KSEL + i;
      mapping[gnode] = r;
      perm_ws[r] = gnode;
      ssel_ws[r] = s[i];
      out[PERM_OFF  + r] = (float)gnode;
      out[SCORE_OFF + r] = s[i];
      out[BATCH_OFF + r] = (float)b;
    } else {
      mapping[gnode] = -1;
    }
  }
}

// ============================================================
// Kernel 3: x_out[r] = x[perm[r]] * score_sel[r]  (wave-per-row)
// ============================================================
__global__ __launch_bounds__(256)
void gather_kernel(const float* __restrict__ x, const int* __restrict__ perm_ws,
                   const float* __restrict__ ssel_ws, float* __restrict__ out) {
  const int wave = threadIdx.x >> 5, lane = threadIdx.x & 31;
  const long r = (long)blockIdx.x * 8 + wave;
  const int src = perm_ws[r];
  const float sc = ssel_ws[r];
  float4 xv = ((const float4*)(x + (long)src * CDIM))[lane];
  float4 o; o.x = xv.x*sc; o.y = xv.y*sc; o.z = xv.z*sc; o.w = xv.w*sc;
  ((float4*)(out + X_OUT_OFF + r * CDIM))[lane] = o;
}

// ============================================================
// Kernel 4: renumber edges through mapping, emit new_ei + mask
// ============================================================
__global__ __launch_bounds__(256)
void edge_kernel(const int* __restrict__ ei, const int* __restrict__ mapping,
                 float* __restrict__ out) {
  const long e = (long)blockIdx.x * 256 + threadIdx.x;
  const int s0 = ei[e];
  const int s1 = ei[ETOT + e];
  const int m0 = mapping[s0];
  const int m1 = mapping[s1];
  const bool keep = (m0 >= 0) && (m1 >= 0);
  out[EI_OFF + e]        = keep ? (float)m0 : -1.0f;
  out[EI_OFF + ETOT + e] = keep ? (float)m1 : -1.0f;
  out[MASK_OFF + e]      = keep ? 1.0f : 0.0f;
}

extern "C" void kernel_launch(void* const* d_in, const int* in_sizes, int n_in,
                              void* d_out, int out_size, void* d_ws, size_t ws_size,
                              hipStream_t stream) {
  const float* x    = (const float*)d_in[0];
  const int*   ei   = (const int*)  d_in[1];
  // d_in[2] = batch : implied by node index (equal-sized graphs), unused
  const float* topo = (const float*)d_in[3];
  const float* Wq   = (const float*)d_in[4];
  const float* Wk   = (const float*)d_in[5];
  const float* pw   = (const float*)d_in[6];
  float* out  = (float*)d_out;
  float* ws_f = (float*)d_ws;
  int*   ws_i = (int*)  d_ws;

  prep_kernel  <<<1,                 512,  0, stream>>>(topo, Wq, Wk, pw, ws_f);
  score_kernel <<<TOTAL / 8,         256,  0, stream>>>(x, ws_f, ws_f + WS_SCORE);
  topk_kernel  <<<BGR,               1024, 0, stream>>>(ws_f + WS_SCORE, ws_i + WS_MAP,
                                                        ws_i + WS_PERM, ws_f + WS_SSEL, out);
  gather_kernel<<<(BGR * KSEL) / 8,  256,  0, stream>>>(x, ws_i + WS_PERM, ws_f + WS_SSEL, out);
  edge_kernel  <<<(int)(ETOT / 256), 256,  0, stream>>>(ei, ws_i + WS_MAP, out);
}